// GHMC_loss_74105365725384
// MI455X (gfx1250) — compile-verified
//
#include <hip/hip_runtime.h>
#include <hip/hip_bf16.h>
#include <stdint.h>

#define BINS   30
#define MMT    0.75f
#define NTOTAL 16777216
#define REPL   64          // histogram replicas: lane-distinct -> conflict-free DS atomics
#define NBLK   2048
#define NTHR   256
#define VEC    2           // b128 loads staged per thread per array per tile

// gfx1250 async global->LDS path (compile-probed via __has_builtin)
#if defined(__has_builtin)
#if __has_builtin(__builtin_amdgcn_global_load_async_to_lds_b128) && \
    __has_builtin(__builtin_amdgcn_s_wait_asynccnt)
#define USE_ASYNC 1
#endif
#endif

// Exact parameter type per hipcc diagnostic: 'vector_size(16) int __device__ *'
typedef int v4i __attribute__((vector_size(4 * sizeof(int))));
typedef __attribute__((address_space(1))) v4i GV4;   // global b128 element
typedef __attribute__((address_space(3))) v4i LV4;   // LDS b128 element

// Per-element math. p = sigmoid(z); both channels share bin b = clip(floor(30*|p-t|)).
// bce0+bce1 = log((1+E)(E+e)) - p - [1 - p - t + 2pt],  E = e^p.
__device__ __forceinline__ void process4(float4 pv, int4 tv,
                                         uint32_t* __restrict__ hcnt,
                                         float* __restrict__ hsum, int rep) {
  float zs[4] = {pv.x, pv.y, pv.z, pv.w};
  int   ts[4] = {tv.x, tv.y, tv.z, tv.w};
#pragma unroll
  for (int k = 0; k < 4; ++k) {
    float z = zs[k];
    float t = (float)ts[k];
    float p = __builtin_amdgcn_rcpf(1.0f + __expf(-z));   // v_rcp_f32 + v_exp_f32
    float g = fabsf(p - t);
    int b = (int)(g * (float)BINS);
    b = (b > BINS - 1) ? (BINS - 1) : (b < 0 ? 0 : b);
    float E   = __expf(p);
    float dot = 1.0f - p - t + 2.0f * p * t;
    float bce = __logf((1.0f + E) * (E + 2.71828182845904523f)) - p - dot;
    atomicAdd(&hcnt[b * REPL + rep], 1u);      // ds_add_u32 (no-return)
    atomicAdd(&hsum[b * REPL + rep], bce);     // ds_add_f32 (no-return)
  }
}

__global__ __launch_bounds__(NTHR) void ghm_main(const float* __restrict__ pred,
                                                 const int* __restrict__ targ,
                                                 uint32_t* __restrict__ g_cnt,
                                                 float* __restrict__ g_sum) {
  __shared__ uint32_t hcnt[BINS * REPL];
  __shared__ float    hsum[BINS * REPL];
  const int tid = threadIdx.x;
  for (int i = tid; i < BINS * REPL; i += NTHR) { hcnt[i] = 0u; hsum[i] = 0.0f; }
  __syncthreads();
  const int rep = tid & (REPL - 1);

#if USE_ASYNC
  // Double-buffered tiles staged through LDS by the gfx1250 async-DMA path
  // (ASYNCcnt-tracked): the copy of tile i+1 overlaps the exp/log-heavy
  // compute of tile i. 2048 elements per tile = 8 elements/thread/barrier.
  __shared__ float4 sp[2][NTHR * VEC];
  __shared__ int4   st[2][NTHR * VEC];
  const int nTiles = NTOTAL / (NTHR * 4 * VEC);    // 8192 tiles of 2048 elems
  const float4* gp4 = (const float4*)pred;
  const int4*   gt4 = (const int4*)targ;

  auto issueTile = [&](int tile, int buf) {
#pragma unroll
    for (int v = 0; v < VEC; ++v) {
      const float4* gp = gp4 + (size_t)tile * (NTHR * VEC) + v * NTHR + tid;
      const int4*   gt = gt4 + (size_t)tile * (NTHR * VEC) + v * NTHR + tid;
      __builtin_amdgcn_global_load_async_to_lds_b128(
          (GV4*)(uintptr_t)gp, (LV4*)(uintptr_t)&sp[buf][v * NTHR + tid], 0, 0);
      __builtin_amdgcn_global_load_async_to_lds_b128(
          (GV4*)(uintptr_t)gt, (LV4*)(uintptr_t)&st[buf][v * NTHR + tid], 0, 0);
    }
  };

  int tile = blockIdx.x;
  if (tile < nTiles) issueTile(tile, 0);
  int buf = 0;
  for (; tile < nTiles; tile += gridDim.x, buf ^= 1) {
    __builtin_amdgcn_s_wait_asynccnt(0);   // this wave's copies for 'buf' landed
    __syncthreads();                       // all waves' copies landed; prev reads of buf^1 done
    int nt = tile + gridDim.x;
    if (nt < nTiles) issueTile(nt, buf ^ 1);
#pragma unroll
    for (int v = 0; v < VEC; ++v)
      process4(sp[buf][v * NTHR + tid], st[buf][v * NTHR + tid], hcnt, hsum, rep);
  }
#else
  // Direct streaming path: b128 loads + gfx1250 global_prefetch_b8 one stride ahead.
  const float4* p4 = (const float4*)pred;
  const int4*   t4 = (const int4*)targ;
  const int n4 = NTOTAL / 4;
  const int stride = gridDim.x * NTHR;
  for (int idx = blockIdx.x * NTHR + tid; idx < n4; idx += stride) {
    if (idx + stride < n4) {
      __builtin_prefetch(&p4[idx + stride], 0, 0);
      __builtin_prefetch(&t4[idx + stride], 0, 0);
    }
    process4(p4[idx], t4[idx], hcnt, hsum, rep);
  }
#endif

  __syncthreads();
  // Tree-reduce the REPL replicas per bin (constant shifts after full unroll).
#pragma unroll
  for (int sh = 5; sh >= 0; --sh) {
    const int s = 1 << sh;
    for (int i = tid; i < BINS * s; i += NTHR) {
      int b = i >> sh, r = i & (s - 1);
      hcnt[b * REPL + r] += hcnt[b * REPL + r + s];
      hsum[b * REPL + r] += hsum[b * REPL + r + s];
    }
    __syncthreads();
  }
  if (tid < BINS) {
    atomicAdd(&g_cnt[tid], hcnt[tid * REPL]);
    atomicAdd(&g_sum[tid], hsum[tid * REPL]);
  }
}

__global__ void ghm_init(uint32_t* g_cnt, float* g_sum) {
  int t = threadIdx.x;
  if (t < BINS) { g_cnt[t] = 0u; g_sum[t] = 0.0f; }
}

__global__ void ghm_final(const uint32_t* __restrict__ g_cnt,
                          const float* __restrict__ g_sum,
                          const float* __restrict__ acc_sum,
                          float* __restrict__ out) {
  if (threadIdx.x != 0 || blockIdx.x != 0) return;
  const float total = 2.0f * (float)NTOTAL;      // x.size = 2N (exact: 2^25)
  int n = 0;
  for (int b = 0; b < BINS; ++b) n += (g_cnt[b] > 0u) ? 1 : 0;
  float nf = (float)(n > 0 ? n : 1);
  float loss = 0.0f;
  for (int b = 0; b < BINS; ++b) {
    uint32_t c = g_cnt[b];
    if (c > 0u) {
      float cf     = 2.0f * (float)c;                        // both channels
      float newacc = MMT * acc_sum[b] + (1.0f - MMT) * cf;   // momentum update
      float w      = (total / fmaxf(newacc, 1e-12f)) / nf;
      loss += w * g_sum[b];
    }
  }
  out[0] = loss / total;   // * LOSS_WEIGHT (1.0)
}

extern "C" void kernel_launch(void* const* d_in, const int* in_sizes, int n_in,
                              void* d_out, int out_size, void* d_ws, size_t ws_size,
                              hipStream_t stream) {
  (void)in_sizes; (void)n_in; (void)out_size; (void)ws_size;
  const float* pred = (const float*)d_in[0];   // [N] f32 logits
  const float* acc  = (const float*)d_in[1];   // [30] f32 momentum state
  const int*   targ = (const int*)d_in[2];     // [N] i32 {0,1}
  uint32_t* g_cnt = (uint32_t*)d_ws;
  float*    g_sum = (float*)((char*)d_ws + BINS * sizeof(uint32_t));
  float*    out   = (float*)d_out;

  ghm_init <<<1, 32, 0, stream>>>(g_cnt, g_sum);
  ghm_main <<<NBLK, NTHR, 0, stream>>>(pred, targ, g_cnt, g_sum);
  ghm_final<<<1, 32, 0, stream>>>(g_cnt, g_sum, acc, out);
}